// CustomSTFT_10943576670895
// MI455X (gfx1250) — compile-verified
//
#include <hip/hip_runtime.h>
#include <math.h>
#include <stdint.h>

// ---------------- problem constants ----------------
#define N_FFT   800
#define HOP     200
#define FREQ    401
#define BATCH   8
#define TLEN    480000
#define PADH    400          // N_FFT/2
#define NFRAMES 2401
#define ELEN    480800       // (NFRAMES-1)*HOP + N_FFT
#define FTILES  151          // ceil(NFRAMES/16)

// ---------------- tiling constants -----------------
#define NT_FWD    26         // ceil(401/16)
#define NT_INV    50         // 800/16
#define K_INV     404        // 401 rounded up to multiple of 4
#define RP_STRIDE 418        // rp/ip rows, stride mod 64 = 34 -> conflict-free

// frames LDS layout produced by TDM padding (interval=4 DWORDs, amount=1 DWORD):
//   laddr(idx) = idx + (idx >> 2)   with idx = f*800 + n
#define FRAMES_FLOATS 16000  // 16*800*5/4
#define LDS_RP_OFF    FRAMES_FLOATS                // 16000
#define LDS_IP_OFF    (LDS_RP_OFF + 16*RP_STRIDE)  // 22688
#define LDS_FLOATS    (LDS_IP_OFF + 16*RP_STRIDE)  // 29376 floats = 117504 B

// ---------------- workspace layout (floats) --------
#define WS_ACC_OFF   0
#define WS_WDR_OFF   (BATCH * ELEN)                 // 3846400
#define WS_WDI_OFF   (WS_WDR_OFF + N_FFT * FREQ)    // +320800
#define WS_WIC_OFF   (WS_WDI_OFF + N_FFT * FREQ)
#define WS_WIS_OFF   (WS_WIC_OFF + N_FFT * FREQ)

typedef float v2f  __attribute__((ext_vector_type(2)));
typedef float v8f  __attribute__((ext_vector_type(8)));
typedef unsigned int u32x4 __attribute__((ext_vector_type(4)));
typedef int   i32x8 __attribute__((ext_vector_type(8)));
typedef int   i32x4 __attribute__((ext_vector_type(4)));

__device__ __forceinline__ int reflect_idx(int j) {
    j = (j < 0) ? -j : j;
    j = (j >= TLEN) ? (2 * TLEN - 2 - j) : j;
    return j;
}

// FP32 WMMA: D(16x16) = A(16x4) * B(4x16) + C   (V_WMMA_F32_16X16X4_F32)
__device__ __forceinline__ v8f wmma_f32(v2f a, v2f b, v8f c) {
    return __builtin_amdgcn_wmma_f32_16x16x4_f32(
        false, a, false, b, (short)0, c, false, false);
}

// ---------------- kernel 0a: zero the overlap-add accumulator ---------------
__global__ void zero_acc_kernel(float* __restrict__ acc, int n) {
    int i = blockIdx.x * blockDim.x + threadIdx.x;
    if (i < n) acc[i] = 0.0f;
}

// ---------------- kernel 0b: fold Hann window into forward DFT matrices -----
__global__ void fold_fwd_kernel(const float* __restrict__ dft_real,
                                const float* __restrict__ dft_imag,
                                const float* __restrict__ window,
                                float* __restrict__ wdr, float* __restrict__ wdi) {
    int i = blockIdx.x * blockDim.x + threadIdx.x;
    if (i >= N_FFT * FREQ) return;
    const float w = window[i / FREQ];          // row n of [800][401]
    wdr[i] = dft_real[i] * w;
    wdi[i] = dft_imag[i] * w;
}

// ---------------- kernel 0c: fold Hann window into inverse DFT matrices -----
__global__ void fold_inv_kernel(const float* __restrict__ idft_cos,
                                const float* __restrict__ idft_sin,
                                const float* __restrict__ window,
                                float* __restrict__ wic, float* __restrict__ wis) {
    int i = blockIdx.x * blockDim.x + threadIdx.x;
    if (i >= FREQ * N_FFT) return;
    const float w = window[i % N_FFT];         // col n of [401][800]
    wic[i] = idft_cos[i] * w;
    wis[i] = idft_sin[i] * w;
}

// ---------------- kernel 1: fused STFT -> mag/phase -> ISTFT -> OLA ---------
__global__ void __launch_bounds__(256)
stft_istft_kernel(const float* __restrict__ x,
                  const float* __restrict__ wdr,   // win-folded dft_real [800][401]
                  const float* __restrict__ wdi,   // win-folded dft_imag [800][401]
                  const float* __restrict__ wic,   // win-folded idft_cos [401][800]
                  const float* __restrict__ wis,   // win-folded idft_sin [401][800]
                  float* __restrict__ acc)         // [BATCH][ELEN]
{
    extern __shared__ float lds[];
    float* frames = lds;                    // padded raw frames, laddr(idx)
    float* rp_sh  = lds + LDS_RP_OFF;       // [16][RP_STRIDE]
    float* ip_sh  = lds + LDS_IP_OFF;       // [16][RP_STRIDE]

    const int tile = blockIdx.x;            // frame tile (16 frames)
    const int b    = blockIdx.y;            // batch
    const int tid  = threadIdx.x;
    const int lane = tid & 31;
    const int wv   = tid >> 5;              // wave id 0..7
    const int mrow = lane & 15;             // A-matrix M row / B-matrix N col
    const int half = lane >> 4;             // selects K pair (ISA 7.12.2 layout)

    // ---- Phase 1: stage 16 raw frames into padded LDS layout ----
    // Interior tiles: no reflection, rows are a strided 2D tile of x -> TDM.
    const bool interior = (tile >= 1) && (tile <= 148);
    if (interior) {
        if (wv == 0) {
            const uint64_t gaddr =
                (uint64_t)(uintptr_t)(x + (size_t)b * TLEN + tile * (16 * HOP) - PADH);
            const uint32_t ldsa = (uint32_t)(uintptr_t)frames;  // low 32 = LDS offset
            u32x4 g0;
            g0[0] = 1u;                                         // count=1, user D#
            g0[1] = ldsa;                                       // lds_addr (bytes)
            g0[2] = (uint32_t)(gaddr & 0xFFFFFFFFu);            // global_addr[31:0]
            g0[3] = (uint32_t)((gaddr >> 32) & 0x1FFFFFFu)      // global_addr[56:32]
                  | (2u << 30);                                 // type=2 (image)
            i32x8 g1;
            g1[0] = (2 << 16)    // data_size = 4B
                  | (1 << 20)    // pad_enable
                  | (1 << 22)    // pad_interval code 1 = every 4 DWORDs
                  | (0 << 25);   // pad_amount  code 0 = 1 DWORD
            g1[1] = (int)((N_FFT & 0xFFFF) << 16);   // tensor_dim0[15:0] @ [63:48]
            g1[2] = (int)(16u << 16);                // dim0 hi=0 | tensor_dim1=16 @ [95:80]
            g1[3] = (int)((unsigned)N_FFT << 16);    // dim1 hi=0 | tile_dim0=800 @ [127:112]
            g1[4] = 16;                              // tile_dim1=16, tile_dim2=0
            g1[5] = HOP;                             // tensor_dim0_stride[31:0] = 200
            g1[6] = 0;                               // stride hi | dim1_stride lo
            g1[7] = 0;
            i32x4 z4 = {0, 0, 0, 0};
            i32x8 z8 = {0, 0, 0, 0, 0, 0, 0, 0};
            __builtin_amdgcn_tensor_load_to_lds(g0, g1, z4, z4, z8, 0);
            __builtin_amdgcn_s_wait_tensorcnt(0);
        }
    } else {
        // edge tiles (0, 149, 150): reflection and/or pad frames, scalar staging
        for (int i = tid; i < 16 * N_FFT; i += 256) {
            const int f = i / N_FFT;
            const int n = i - f * N_FFT;
            const int F = tile * 16 + f;
            float v = 0.0f;
            if (F < NFRAMES) {
                const int j = reflect_idx(F * HOP + n - PADH);
                v = x[(size_t)b * TLEN + j];
            }
            frames[i + (i >> 2)] = v;       // padded layout
        }
    }
    __syncthreads();

    // ---- Phase 2: forward GEMM (re, im), then mag/phase rescale -> rp/ip ----
    for (int nt = wv; nt < NT_FWD; nt += 8) {
        const int col  = nt * 16 + mrow;
        const int colc = (col < FREQ) ? col : (FREQ - 1);   // clamp, zeroed later
        v8f re_acc = {0.f,0.f,0.f,0.f,0.f,0.f,0.f,0.f};
        v8f im_acc = {0.f,0.f,0.f,0.f,0.f,0.f,0.f,0.f};
        for (int k = 0; k < N_FFT; k += 4) {
            const int ka   = k + 2 * half;                  // this lane's K pair
            const int aidx = 1000 * mrow + ka + (ka >> 2);  // padded frames layout
            v2f a;  a.x  = frames[aidx];              a.y  = frames[aidx + 1];
            v2f br; br.x = wdr[ka * FREQ + colc];     br.y = wdr[(ka + 1) * FREQ + colc];
            v2f bi; bi.x = wdi[ka * FREQ + colc];     bi.y = wdi[(ka + 1) * FREQ + colc];
            re_acc = wmma_f32(a, br, re_acc);
            im_acc = wmma_f32(a, bi, im_acc);
        }
        // D layout: lane holds (M = v + 8*half, N = col) for v = 0..7
        #pragma unroll
        for (int v = 0; v < 8; ++v) {
            const int mo = v + 8 * half;
            float re = re_acc[v], im = im_acc[v];
            float h   = re * re + im * im;
            float mag = __builtin_amdgcn_sqrtf(h + 1e-14f);
            float s   = mag * __builtin_amdgcn_rsqf(h);     // inf/NaN if h==0
            float rp  = (h > 0.0f) ? re * s : mag;          // atan2(0,0)=0 path
            float ip  = (h > 0.0f) ? im * s : 0.0f;
            if (col >= FREQ) { rp = 0.0f; ip = 0.0f; }      // zero-pad K for inverse
            rp_sh[mo * RP_STRIDE + col] = rp;
            ip_sh[mo * RP_STRIDE + col] = ip;
        }
    }
    __syncthreads();

    // ---- Phase 3: inverse GEMM (window pre-folded), OLA via global atomics ----
    for (int nt = wv; nt < NT_INV; nt += 8) {
        const int n = nt * 16 + mrow;          // output time sample within frame
        v8f cacc = {0.f,0.f,0.f,0.f,0.f,0.f,0.f,0.f};
        v8f sacc = {0.f,0.f,0.f,0.f,0.f,0.f,0.f,0.f};
        const float* rrow = rp_sh + mrow * RP_STRIDE;
        const float* irow = ip_sh + mrow * RP_STRIDE;
        for (int k = 0; k < K_INV; k += 4) {
            const int ka = k + 2 * half;
            v2f ar; ar.x = rrow[ka]; ar.y = rrow[ka + 1];
            v2f ai; ai.x = irow[ka]; ai.y = irow[ka + 1];
            const int k0 = (ka     < FREQ) ? ka     : (FREQ - 1);  // A is 0 there
            const int k1 = (ka + 1 < FREQ) ? ka + 1 : (FREQ - 1);
            v2f bc; bc.x = wic[k0 * N_FFT + n]; bc.y = wic[k1 * N_FFT + n];
            v2f bs; bs.x = wis[k0 * N_FFT + n]; bs.y = wis[k1 * N_FFT + n];
            cacc = wmma_f32(ar, bc, cacc);
            sacc = wmma_f32(ai, bs, sacc);
        }
        #pragma unroll
        for (int v = 0; v < 8; ++v) {
            const int mo = v + 8 * half;
            const int F  = tile * 16 + mo;
            if (F < NFRAMES) {
                atomicAdd(&acc[(size_t)b * ELEN + F * HOP + n], cacc[v] - sacc[v]);
            }
        }
    }
}

// ---------------- kernel 2: normalize by overlap-added win^2, crop ----------
__global__ void finalize_kernel(const float* __restrict__ acc,
                                const float* __restrict__ window,
                                float* __restrict__ out)
{
    int i = blockIdx.x * blockDim.x + threadIdx.x;
    if (i >= BATCH * TLEN) return;
    const int b = i / TLEN;
    const int t = i - b * TLEN;
    const int p = t + PADH;
    int f_lo = (p > 799) ? ((p - 600) / 200) : 0;      // ceil((p-799)/200)
    int f_hi = p / HOP; if (f_hi > NFRAMES - 1) f_hi = NFRAMES - 1;
    float norm = 0.0f;
    for (int f = f_lo; f <= f_hi; ++f) {
        const float w = window[p - HOP * f];
        norm += w * w;
    }
    out[i] = acc[(size_t)b * ELEN + p] * __builtin_amdgcn_rcpf(norm + 1e-14f);
}

// ---------------- host launcher ----------------
extern "C" void kernel_launch(void* const* d_in, const int* in_sizes, int n_in,
                              void* d_out, int out_size, void* d_ws, size_t ws_size,
                              hipStream_t stream) {
    (void)in_sizes; (void)n_in; (void)out_size; (void)ws_size;
    const float* x        = (const float*)d_in[0];
    const float* window   = (const float*)d_in[1];
    const float* dft_real = (const float*)d_in[2];
    const float* dft_imag = (const float*)d_in[3];
    const float* idft_cos = (const float*)d_in[4];
    const float* idft_sin = (const float*)d_in[5];
    float* out = (float*)d_out;

    float* ws  = (float*)d_ws;
    float* acc = ws + WS_ACC_OFF;     // BATCH*ELEN
    float* wdr = ws + WS_WDR_OFF;     // 800*401
    float* wdi = ws + WS_WDI_OFF;
    float* wic = ws + WS_WIC_OFF;     // 401*800
    float* wis = ws + WS_WIS_OFF;

    const int    nacc     = BATCH * ELEN;
    const int    nmat     = N_FFT * FREQ;
    const size_t ldsBytes = (size_t)LDS_FLOATS * sizeof(float);  // 117504 B

    // Opt in to >64KB dynamic LDS (WGP has 320KB). Host-side, capture-safe.
    (void)hipFuncSetAttribute((const void*)stft_istft_kernel,
                              hipFuncAttributeMaxDynamicSharedMemorySize,
                              (int)ldsBytes);

    zero_acc_kernel<<<(nacc + 255) / 256, 256, 0, stream>>>(acc, nacc);
    fold_fwd_kernel<<<(nmat + 255) / 256, 256, 0, stream>>>(dft_real, dft_imag,
                                                            window, wdr, wdi);
    fold_inv_kernel<<<(nmat + 255) / 256, 256, 0, stream>>>(idft_cos, idft_sin,
                                                            window, wic, wis);

    stft_istft_kernel<<<dim3(FTILES, BATCH), 256, ldsBytes, stream>>>(
        x, wdr, wdi, wic, wis, acc);

    const int nout = BATCH * TLEN;
    finalize_kernel<<<(nout + 255) / 256, 256, 0, stream>>>(acc, window, out);
}